// PKC2D_Sample_5016521802446
// MI455X (gfx1250) — compile-verified
//
#include <hip/hip_runtime.h>
#include <hip/hip_bf16.h>

typedef float v2f __attribute__((ext_vector_type(2)));
typedef float v8f __attribute__((ext_vector_type(8)));

#define IMG_W   256
#define IMG_HW  65536        // 256*256
#define CH      16
#define NRING   40
#define NSEL    16
#define BLK     128          // threads per block = pixels per block

// Ring offsets of the 11x11 window border, in exact _gen_prf_grid order:
//  n in [0,11)  : dx=-5,    dy=n-5      (top row)
//  n in [11,20) : dx=n-15,  dy=+5       (right col, rows -4..4)
//  n in [20,31) : dx=+5,    dy=n-25     (bottom row)
//  n in [31,40) : dx=n-35,  dy=-5       (left col, rows -4..4)
__device__ __forceinline__ void ring_offset(int n, int& dx, int& dy) {
    if (n < 11)      { dx = -5;     dy = n - 5;  }
    else if (n < 20) { dx = n - 15; dy = 5;      }
    else if (n < 31) { dx = 5;      dy = n - 25; }
    else             { dx = n - 35; dy = -5;     }
}

extern "C" __global__ __launch_bounds__(BLK)
void pkc2d_fused_wmma(const float* __restrict__ x,
                      const float* __restrict__ conv_w,
                      const float* __restrict__ conv_b,
                      float* __restrict__ out) {
    // Dynamic LDS: [0, 32768) floats  = A data, layout [mtile][kpair][m][2]
    //              [32768, 40960)     = weights, layout [kpair][o][2]
    extern __shared__ float lds[];
    float* A_lds = lds;
    float* W_lds = lds + 32768;

    const int tid = threadIdx.x;
    const int p   = blockIdx.x * BLK + tid;     // global pixel id
    const int j   = p & 255;
    const int i   = (p >> 8) & 255;
    const int bb  = p >> 16;
    const float* xb = x + (size_t)bb * CH * IMG_HW;

    // ---- stage weights into LDS: W_lds[(k>>1)*64 + o*2 + (k&1)] = W[o][k]
    for (int t = tid; t < 32 * 256; t += BLK) {
        int o = t & 31;
        int k = t >> 5;
        W_lds[(k >> 1) * 64 + o * 2 + (k & 1)] = conv_w[o * 256 + k];
    }

    // ---- phase 1: center pixel, 40 similarity dots (sigmoid/mean skipped: monotone)
    float xc[CH];
    #pragma unroll
    for (int ch = 0; ch < CH; ++ch)
        xc[ch] = xb[ch * IMG_HW + i * IMG_W + j];

    float s[NRING];
    #pragma unroll
    for (int n = 0; n < NRING; ++n) {
        int dx, dy;
        ring_offset(n, dx, dy);                 // compile-time after unroll
        int ii = i + dx, jj = j + dy;
        float dot = 0.f;
        if ((unsigned)ii < 256u && (unsigned)jj < 256u) {  // zero padding -> dot 0
            #pragma unroll
            for (int ch = 0; ch < CH; ++ch)
                dot += xc[ch] * xb[ch * IMG_HW + ii * IMG_W + jj];
        }
        s[n] = dot;
    }

    // ---- rank-n selection (16 smallest ascending, stable by index on ties),
    //      gather selected neighbors, stage A row (x - x_r) into LDS.
    const int m  = tid & 15;                    // position within M-tile
    const int mt = tid >> 4;                    // M-tile within block (0..7)
    float* Arow  = A_lds + mt * 4096 + m * 2;

    float pv = -3.4e38f;
    int   pi = -1;
    #pragma unroll
    for (int n = 0; n < NSEL; ++n) {
        float bv = 3.4e38f;
        int   bi = 64;
        #pragma unroll
        for (int q = 0; q < NRING; ++q) {
            // candidate must be lexicographically greater than previous pick
            bool gt = (s[q] > pv) || (s[q] == pv && q > pi);
            bool lt = gt && ((s[q] < bv) || (s[q] == bv && q < bi));
            if (lt) { bv = s[q]; bi = q; }
        }
        int dx, dy;
        ring_offset(bi, dx, dy);
        int ii = i + dx, jj = j + dy;
        bool ok = (unsigned)ii < 256u && (unsigned)jj < 256u;
        #pragma unroll
        for (int ch = 0; ch < CH; ++ch) {
            float xr = ok ? xb[ch * IMG_HW + ii * IMG_W + jj] : 0.f;
            int k = ch * 16 + n;                // matches W[o,ch,k1,k2] flattening
            Arow[(k >> 1) * 32 + (k & 1)] = xc[ch] - xr;
        }
        pv = bv; pi = bi;
    }

    __syncthreads();

    // ---- phase 2: per-wave WMMA GEMM  D[o, pos] = sum_k W[o,k] * A[k,pos]
    // A operand (WMMA-A, 16x4 f32): lane l -> M=o=l%16, K = kbase + 2*(l/16) + v
    // B operand (WMMA-B, 4x16 f32): lane l -> N=pos=l%16, K = kbase + 2*(l/16) + v
    const int lane = tid & 31;
    const int wv   = tid >> 5;                  // wave id 0..3
    const int lm   = lane & 15;
    const int kh   = lane >> 4;

    for (int mt2 = 2 * wv; mt2 < 2 * wv + 2; ++mt2) {
        v8f acc0 = {};                          // o in [0,16)
        v8f acc1 = {};                          // o in [16,32)
        for (int k = 0; k < 256; k += 4) {
            int kp = (k >> 1) + kh;
            v2f bfrag = *(const v2f*)(A_lds + mt2 * 4096 + kp * 32 + lm * 2);
            v2f a0    = *(const v2f*)(W_lds + kp * 64 + lm * 2);
            v2f a1    = *(const v2f*)(W_lds + kp * 64 + (lm + 16) * 2);
            acc0 = __builtin_amdgcn_wmma_f32_16x16x4_f32(
                       false, a0, false, bfrag, (short)0, acc0, false, false);
            acc1 = __builtin_amdgcn_wmma_f32_16x16x4_f32(
                       false, a1, false, bfrag, (short)0, acc1, false, false);
        }
        // D layout: lanes 0-15: M=r, N=lane; lanes 16-31: M=r+8, N=lane-16
        int pbase = blockIdx.x * BLK + mt2 * 16 + lm;   // pixel this lane writes
        int bbo   = pbase >> 16;
        int sp    = pbase & 65535;                       // i*256 + j
        float* ob = out + (size_t)bbo * 32 * IMG_HW + sp;
        #pragma unroll
        for (int r = 0; r < 8; ++r) {
            int o0 = r + 8 * kh;
            ob[(size_t)o0 * IMG_HW]        = acc0[r] + conv_b[o0];
            ob[(size_t)(o0 + 16) * IMG_HW] = acc1[r] + conv_b[o0 + 16];
        }
    }
}

extern "C" void kernel_launch(void* const* d_in, const int* in_sizes, int n_in,
                              void* d_out, int out_size, void* d_ws, size_t ws_size,
                              hipStream_t stream) {
    const float* x  = (const float*)d_in[0];   // (2,16,256,256) f32
    const float* cw = (const float*)d_in[1];   // (32,16,4,4)    f32
    const float* cb = (const float*)d_in[2];   // (32,)          f32
    float* out = (float*)d_out;                // (2,32,256,256) f32

    const int n_pix = 2 * 256 * 256;           // 131072
    dim3 grid(n_pix / BLK);                    // 1024
    dim3 block(BLK);
    size_t smem = (32768 + 8192) * sizeof(float);  // 160 KB dynamic LDS
    hipLaunchKernelGGL(pkc2d_fused_wmma, grid, block, smem, stream, x, cw, cb, out);
}